// GAT_42193758715953
// MI455X (gfx1250) — compile-verified
//
#include <hip/hip_runtime.h>

#define N_NODES  50000
#define N_EDGES  800000
#define E_TOT    850000   // edges + self loops
#define F        128      // HEADS*HID == IN_CH
#define HEADS    4
#define HID      32
#define OUT_CH   10
#define N_GRAPHS 64
#define NEG_SLOPE 0.2f
#define WSTRIDE  136      // LDS row stride (pad so lane halves hit disjoint banks)

typedef float v2f __attribute__((ext_vector_type(2)));
typedef float v8f __attribute__((ext_vector_type(8)));

// ---------- order-preserving float<->uint map for atomicMax on floats ----------
__device__ __forceinline__ unsigned f2ord(float f) {
  unsigned u = __float_as_uint(f);
  return (u & 0x80000000u) ? ~u : (u | 0x80000000u);
}
__device__ __forceinline__ float ord2f(unsigned o) {
  unsigned u = (o & 0x80000000u) ? (o ^ 0x80000000u) : ~o;
  return __uint_as_float(u);
}

// ---------- H = X @ W  (X:[n x 128], W:[128 x 128] row-major) via f32 WMMA ----------
__global__ __launch_bounds__(256)
void gat_gemm128_wmma(const float* __restrict__ X, const float* __restrict__ W,
                      float* __restrict__ H, int nrows) {
  __shared__ float sW[128 * WSTRIDE];
  for (int i = threadIdx.x; i < 128 * 128; i += 256)
    sW[(i >> 7) * WSTRIDE + (i & 127)] = W[i];
  __syncthreads();

  const int lane = threadIdx.x & 31;
  const int wave = threadIdx.x >> 5;   // 0..7 -> column tile
  const int l16  = lane & 15;
  const int lhi  = lane >> 4;          // 0 or 1
  const int n0   = wave * 16;
  const int rowBlock = blockIdx.x * 128;

  for (int rt = 0; rt < 8; ++rt) {
    const int m0 = rowBlock + rt * 16;          // wave-uniform guard: EXEC stays full
    if (m0 >= nrows) break;
    const float* xrow = X + (size_t)(m0 + l16) * F;
    v8f acc = {};
#pragma unroll
    for (int k = 0; k < 128; k += 4) {
      const int ka = k + 2 * lhi;
      v2f a, b;
      a.x = xrow[ka];                              // A[m][k'] : lane=m+16*(k'>>1), vgpr=k'&1
      a.y = xrow[ka + 1];
      b.x = sW[ka * WSTRIDE + n0 + l16];           // B[k'][n] : lane=n+16*(k'>>1), vgpr=k'&1
      b.y = sW[(ka + 1) * WSTRIDE + n0 + l16];
      acc = __builtin_amdgcn_wmma_f32_16x16x4_f32(false, a, false, b,
                                                  (short)0, acc, false, false);
    }
    float* hp = H + (size_t)(m0 + 8 * lhi) * F + n0 + l16;  // D: M=r+8*lhi, N=l16
#pragma unroll
    for (int r = 0; r < 8; ++r) hp[(size_t)r * F] = acc[r];
  }
}

// ---------- per (node, head): a_src/a_dst = <h[n,h,:], att> ----------
__global__ void gat_attn_coeff(const float* __restrict__ H,
                               const float* __restrict__ att_src,
                               const float* __restrict__ att_dst,
                               float* __restrict__ a_src, float* __restrict__ a_dst) {
  int idx = blockIdx.x * blockDim.x + threadIdx.x;
  if (idx >= N_NODES * HEADS) return;
  const int n = idx >> 2, h = idx & 3;
  const float* hp = H + (size_t)n * F + h * HID;
  const float* as = att_src + h * HID;
  const float* ad = att_dst + h * HID;
  float s = 0.f, d = 0.f;
#pragma unroll
  for (int c = 0; c < HID; ++c) { float v = hp[c]; s += v * as[c]; d += v * ad[c]; }
  a_src[idx] = s; a_dst[idx] = d;
}

__device__ __forceinline__ void edge_nodes(int e, const int* esrc, const int* edst,
                                           int& s, int& d) {
  if (e < N_EDGES) { s = esrc[e]; d = edst[e]; }
  else             { s = e - N_EDGES; d = s; }   // self loop
}

// ---------- pass 1: segment max of leaky-relu logits ----------
__global__ void gat_edge_max(const int* __restrict__ esrc, const int* __restrict__ edst,
                             const float* __restrict__ a_src, const float* __restrict__ a_dst,
                             unsigned* __restrict__ emax) {
  int idx = blockIdx.x * blockDim.x + threadIdx.x;
  if (idx >= E_TOT * HEADS) return;
  const int e = idx >> 2, h = idx & 3;
  int s, d; edge_nodes(e, esrc, edst, s, d);
  float v = a_src[s * HEADS + h] + a_dst[d * HEADS + h];
  v = (v > 0.f) ? v : NEG_SLOPE * v;
  atomicMax(&emax[d * HEADS + h], f2ord(v));
}

// ---------- pass 2: exp weights + segment sum ----------
__global__ void gat_edge_expsum(const int* __restrict__ esrc, const int* __restrict__ edst,
                                const float* __restrict__ a_src, const float* __restrict__ a_dst,
                                const unsigned* __restrict__ emax,
                                float* __restrict__ wbuf, float* __restrict__ denom) {
  int idx = blockIdx.x * blockDim.x + threadIdx.x;
  if (idx >= E_TOT * HEADS) return;
  const int e = idx >> 2, h = idx & 3;
  int s, d; edge_nodes(e, esrc, edst, s, d);
  float v = a_src[s * HEADS + h] + a_dst[d * HEADS + h];
  v = (v > 0.f) ? v : NEG_SLOPE * v;
  float w = __expf(v - ord2f(emax[d * HEADS + h]));
  wbuf[idx] = w;
  atomicAdd(&denom[d * HEADS + h], w);
}

__global__ void gat_recip(const float* __restrict__ denom, float* __restrict__ invden) {
  int idx = blockIdx.x * blockDim.x + threadIdx.x;
  if (idx >= N_NODES * HEADS) return;
  invden[idx] = 1.f / denom[idx];   // every node has a self loop -> denom > 0
}

// ---------- pass 3: AGG[dst] += alpha * H[src]  (one thread per edge-channel) ----------
__global__ void gat_edge_aggregate(const int* __restrict__ esrc, const int* __restrict__ edst,
                                   const float* __restrict__ wbuf,
                                   const float* __restrict__ invden,
                                   const float* __restrict__ H, float* __restrict__ AGG) {
  int idx = blockIdx.x * blockDim.x + threadIdx.x;   // E_TOT*128 = 108.8M < 2^31
  if (idx >= E_TOT * F) return;
  const int e = idx >> 7, c = idx & 127, h = c >> 5;
  int s, d; edge_nodes(e, esrc, edst, s, d);
  const float alpha = wbuf[e * HEADS + h] * invden[d * HEADS + h];
  atomicAdd(&AGG[(size_t)d * F + c], alpha * H[(size_t)s * F + c]);
}

__global__ void gat_bias_relu(float* __restrict__ AGG, const float* __restrict__ bias) {
  int idx = blockIdx.x * blockDim.x + threadIdx.x;
  if (idx >= N_NODES * F) return;
  AGG[idx] = fmaxf(AGG[idx] + bias[idx & 127], 0.f);
}

// ---------- global mean pool ----------
__global__ void gat_pool_feat(const int* __restrict__ batch, const float* __restrict__ Hf,
                              float* __restrict__ sums) {
  int idx = blockIdx.x * blockDim.x + threadIdx.x;
  if (idx >= N_NODES * F) return;
  const int n = idx >> 7, c = idx & 127;
  atomicAdd(&sums[batch[n] * F + c], Hf[idx]);
}

__global__ void gat_pool_cnt(const int* __restrict__ batch, float* __restrict__ cnts) {
  int idx = blockIdx.x * blockDim.x + threadIdx.x;
  if (idx >= N_NODES) return;
  atomicAdd(&cnts[batch[idx]], 1.f);
}

// ---------- final linear on pooled means ----------
__global__ void gat_final_linear(const float* __restrict__ sums, const float* __restrict__ cnts,
                                 const float* __restrict__ wlin, const float* __restrict__ blin,
                                 float* __restrict__ out) {
  int idx = blockIdx.x * blockDim.x + threadIdx.x;
  if (idx >= N_GRAPHS * OUT_CH) return;
  const int g = idx / OUT_CH, o = idx % OUT_CH;
  const float inv = 1.f / fmaxf(cnts[g], 1.f);
  float acc = blin[o];
  for (int c = 0; c < F; ++c) acc += (sums[g * F + c] * inv) * wlin[c * OUT_CH + o];
  out[idx] = acc;
}

extern "C" void kernel_launch(void* const* d_in, const int* in_sizes, int n_in,
                              void* d_out, int out_size, void* d_ws, size_t ws_size,
                              hipStream_t stream) {
  const float* x        = (const float*)d_in[0];
  const int*   esrc     = (const int*)d_in[1];                // edge_index[0]
  const int*   edst     = ((const int*)d_in[1]) + N_EDGES;    // edge_index[1]
  const int*   batch    = (const int*)d_in[2];
  const float* W1       = (const float*)d_in[3];
  const float* att_s1   = (const float*)d_in[4];
  const float* att_d1   = (const float*)d_in[5];
  const float* b1       = (const float*)d_in[6];
  const float* W2       = (const float*)d_in[7];
  const float* att_s2   = (const float*)d_in[8];
  const float* att_d2   = (const float*)d_in[9];
  const float* b2       = (const float*)d_in[10];
  const float* wlin     = (const float*)d_in[11];
  const float* blin     = (const float*)d_in[12];
  float*       out      = (float*)d_out;

  // workspace layout (floats)
  float*    ws     = (float*)d_ws;
  float*    H      = ws;                               // 6,400,000
  float*    AGG    = H      + (size_t)N_NODES * F;     // 6,400,000
  float*    a_src  = AGG    + (size_t)N_NODES * F;     // 200,000
  float*    a_dst  = a_src  + N_NODES * HEADS;         // 200,000
  unsigned* emax   = (unsigned*)(a_dst + N_NODES * HEADS);
  float*    denom  = (float*)(emax + N_NODES * HEADS);
  float*    invden = denom  + N_NODES * HEADS;
  float*    wbuf   = invden + N_NODES * HEADS;         // 3,400,000
  float*    sums   = wbuf   + (size_t)E_TOT * HEADS;   // 8,192
  float*    cnts   = sums   + N_GRAPHS * F;            // 64

  const int T = 256;
  const int gGemm = (N_NODES + 127) / 128;
  const int gNH   = (N_NODES * HEADS + T - 1) / T;
  const int gEH   = (E_TOT * HEADS + T - 1) / T;
  const int gEC   = (E_TOT * F + T - 1) / T;
  const int gNF   = (N_NODES * F + T - 1) / T;
  const int gN    = (N_NODES + T - 1) / T;
  const int gOut  = (N_GRAPHS * OUT_CH + T - 1) / T;

  for (int layer = 0; layer < 2; ++layer) {
    const float* Xin = (layer == 0) ? x : AGG;
    const float* W   = (layer == 0) ? W1 : W2;
    const float* as  = (layer == 0) ? att_s1 : att_s2;
    const float* ad  = (layer == 0) ? att_d1 : att_d2;
    const float* b   = (layer == 0) ? b1 : b2;

    gat_gemm128_wmma<<<gGemm, T, 0, stream>>>(Xin, W, H, N_NODES);
    gat_attn_coeff<<<gNH, T, 0, stream>>>(H, as, ad, a_src, a_dst);

    hipMemsetAsync(emax,  0, sizeof(unsigned) * N_NODES * HEADS, stream);
    hipMemsetAsync(denom, 0, sizeof(float)    * N_NODES * HEADS, stream);
    hipMemsetAsync(AGG,   0, sizeof(float)    * (size_t)N_NODES * F, stream);

    gat_edge_max<<<gEH, T, 0, stream>>>(esrc, edst, a_src, a_dst, emax);
    gat_edge_expsum<<<gEH, T, 0, stream>>>(esrc, edst, a_src, a_dst, emax, wbuf, denom);
    gat_recip<<<gNH, T, 0, stream>>>(denom, invden);
    gat_edge_aggregate<<<gEC, T, 0, stream>>>(esrc, edst, wbuf, invden, H, AGG);
    gat_bias_relu<<<gNF, T, 0, stream>>>(AGG, b);
  }

  hipMemsetAsync(sums, 0, sizeof(float) * N_GRAPHS * F, stream);
  hipMemsetAsync(cnts, 0, sizeof(float) * N_GRAPHS, stream);
  gat_pool_feat<<<gNF, T, 0, stream>>>(batch, AGG, sums);
  gat_pool_cnt<<<gN, T, 0, stream>>>(batch, cnts);
  gat_final_linear<<<gOut, T, 0, stream>>>(sums, cnts, wlin, blin, out);
}